// Group_Quantize_81355270521165
// MI455X (gfx1250) — compile-verified
//
#include <hip/hip_runtime.h>
#include <hip/hip_bf16.h>

// Problem constants (from reference)
#define NG      6          // groups
#define KDIM    40         // vector length per group
#define CB      1024       // codebook size
#define NB      32         // batch
#define TP      2000       // time steps after reshape (8000/4)
#define ROWS    (NB * TP)          // 64000 rows per group
#define TILES_PER_G (ROWS / 16)    // 4000
#define NTILES  (NG * TILES_PER_G) // 24000
#define ZB_N    (60 * 8000)        // 480000 floats per batch elem
#define OUT_Q_ELEMS (NB * ZB_N)    // 15360000
// vq_loss[n] = (0.25/6) * (1/(2000*40)) * sum_{g,t,k} diff^2
#define LOSS_SCALE (0.25f / (6.0f * 80000.0f))

// Packed-B stream: per (g, ct, j2) a 128-float record = 32 lanes x 4 floats,
// holding the WMMA B fragments of k-chunks 2*j2 and 2*j2+1 in lane order.
#define BP_PER_GC   (5 * 128)            // 640 floats per (g, column-tile)
#define BP_PER_G    (64 * BP_PER_GC)     // 40960 floats per group
#define BP_TOTAL    (NG * BP_PER_G)      // 245760 floats (= repacked E)

typedef __attribute__((ext_vector_type(2))) float v2f;
typedef __attribute__((ext_vector_type(4))) float v4f;
typedef __attribute__((ext_vector_type(8))) float v8f;

// Kernel 1: repack E into WMMA lane order, precompute ||E_c||^2, zero losses.
__global__ void vq_prep_kernel(const float* __restrict__ emb,
                               float* __restrict__ bpack,
                               float* __restrict__ e2,
                               float* __restrict__ loss) {
    const int i = blockIdx.x * blockDim.x + threadIdx.x;
    if (i < BP_TOTAL) {
        const int e    = i & 3;                 // element within lane's float4
        const int lane = (i >> 2) & 31;
        const int j2   = (i >> 7) % 5;          // k-chunk pair
        const int r    = i / BP_PER_GC;         // g*64 + ct
        const int ct   = r & 63;
        const int g    = r >> 6;
        const int j    = 2 * j2 + (e >> 1);     // k-chunk 0..9
        const int comp = e & 1;                 // .x/.y within fragment
        const int koff = (lane >> 4) << 1;      // lanes 16-31 carry K+2,K+3
        const int k    = 4 * j + koff + comp;
        const int col  = (ct << 4) + (lane & 15);
        bpack[i] = emb[(size_t)g * KDIM * CB + (size_t)k * CB + col];
    }
    if (i < NG * CB) {
        const int g = i >> 10;
        const int c = i & (CB - 1);
        const float* p = emb + (size_t)g * KDIM * CB + c;
        float s = 0.f;
        #pragma unroll
        for (int k = 0; k < KDIM; ++k) {
            const float ev = p[(size_t)k * CB];
            s += ev * ev;
        }
        e2[i] = s;
    }
    if (i < NB) loss[i] = 0.f;
}

// Kernel 2: one wave32 per 16-row tile. Distances via V_WMMA_F32_16X16X4_F32:
//   D = (-2x) * E + C,   C initialized to ||E_c||^2  =>  D = ||E_c||^2 - 2 x.E
__global__ void __launch_bounds__(256)
vq_main_kernel(const float* __restrict__ z,
               const float* __restrict__ emb,
               const float* __restrict__ bpack,
               const float* __restrict__ e2,
               float* __restrict__ out,
               float* __restrict__ loss) {
    const int wave = threadIdx.x >> 5;
    const int lane = threadIdx.x & 31;
    const int tile = blockIdx.x * 8 + wave;        // 0..NTILES-1 exactly

    const int g  = tile / TILES_PER_G;
    const int tg = tile - g * TILES_PER_G;         // tile within group
    const int n  = tg / (TP / 16);                 // batch index (125 tiles per n)
    const int t0 = (tg - n * (TP / 16)) << 4;      // time offset of this 16-row tile

    // x[m,k] = z[n*480000 + (g*40+k)*2000 + t0 + m]
    const float* __restrict__ zg  = z     + (size_t)n * ZB_N + (size_t)g * KDIM * TP + t0;
    const float* __restrict__ Eg  = emb   + (size_t)g * KDIM * CB;
    const float* __restrict__ e2g = e2    + (g << 10);
    const float* __restrict__ Bg  = bpack + (size_t)g * BP_PER_G + (lane << 2);
    float*       __restrict__ og  = out   + (size_t)n * ZB_N + (size_t)g * KDIM * TP + t0;

    const int mh   = lane & 15;          // M (for A/C) or N (for B) within tile
    const int koff = (lane >> 4) << 1;   // lanes 0-15: K=k0,k0+1 ; lanes 16-31: K=k0+2,k0+3

    // ---- Load A = -2*x, 10 chunks of 16x4 (ISA 32-bit A layout) ----
    v2f A[10];
    #pragma unroll
    for (int j = 0; j < 10; ++j) {
        const int k0 = 4 * j + koff;
        A[j].x = -2.0f * zg[(size_t)(k0    ) * TP + mh];
        A[j].y = -2.0f * zg[(size_t)(k0 + 1) * TP + mh];
    }

    float bestv[8];
    int   besti[8];
    #pragma unroll
    for (int v = 0; v < 8; ++v) { bestv[v] = 3.0e38f; besti[v] = 0; }

    // ---- Sweep 64 column tiles of the codebook ----
    for (int ct = 0; ct < 64; ++ct) {
        const int n0 = ct << 4;
        const float c0 = e2g[n0 + mh];                 // ||E_c||^2 for this lane's column
        v8f acc = { c0, c0, c0, c0, c0, c0, c0, c0 };  // C init = ||E||^2
        const float* __restrict__ bt = Bg + ct * BP_PER_GC;
        #pragma unroll
        for (int j2 = 0; j2 < 5; ++j2) {
            const v4f b = *(const v4f*)(bt + j2 * 128);   // global_load_b128
            acc = __builtin_amdgcn_wmma_f32_16x16x4_f32(
                false, A[2 * j2],     false, __builtin_shufflevector(b, b, 0, 1),
                (short)0, acc, false, false);
            acc = __builtin_amdgcn_wmma_f32_16x16x4_f32(
                false, A[2 * j2 + 1], false, __builtin_shufflevector(b, b, 2, 3),
                (short)0, acc, false, false);
        }
        // branch-free argmin update (v_cndmask; EXEC stays all-1s for WMMA)
        const int cidx = n0 + mh;
        #pragma unroll
        for (int v = 0; v < 8; ++v) {
            const float d = acc[v];
            const bool c = d < bestv[v];
            bestv[v] = c ? d    : bestv[v];
            besti[v] = c ? cidx : besti[v];
        }
    }

    // ---- Reduce argmin across the 16 lanes of each half (rows 0-7 / 8-15) ----
    #pragma unroll
    for (int off = 1; off <= 8; off <<= 1) {
        #pragma unroll
        for (int v = 0; v < 8; ++v) {
            const float ov = __shfl_xor(bestv[v], off, 32);
            const int   oi = __shfl_xor(besti[v], off, 32);
            const bool c = (ov < bestv[v]) || (ov == bestv[v] && oi < besti[v]);
            bestv[v] = c ? ov : bestv[v];
            besti[v] = c ? oi : besti[v];
        }
    }
    // lanes 0-15 now hold rows 0-7 in slots 0-7; lanes 16-31 hold rows 8-15.
    // Distribute: lane l handles row m = l&15; the two halves split the k range.
    int my_idx = 0;
    #pragma unroll
    for (int v = 0; v < 8; ++v) {
        const int lo = __shfl(besti[v], 0,  32);  // row v
        const int hi = __shfl(besti[v], 16, 32);  // row v+8
        if (mh == v)     my_idx = lo;
        if (mh == v + 8) my_idx = hi;
    }

    // ---- Writeback quantized vectors + commitment loss ----
    const float* __restrict__ Ecol = Eg + my_idx;
    const int kbase = (lane >> 4) * (KDIM / 2);   // half 0: k 0..19, half 1: k 20..39
    float lsum = 0.f;
    #pragma unroll
    for (int kk = 0; kk < KDIM / 2; ++kk) {
        const int k = kbase + kk;
        const float e  = Ecol[(size_t)k * CB];
        const float xv = zg[(size_t)k * TP + mh];
        const float d  = e - xv;
        lsum += d * d;
        og[(size_t)k * TP + mh] = e;              // coalesced per half-wave
    }
    #pragma unroll
    for (int off = 16; off >= 1; off >>= 1)
        lsum += __shfl_xor(lsum, off, 32);
    if (lane == 0)
        atomicAdd(loss + n, lsum * LOSS_SCALE);
}

extern "C" void kernel_launch(void* const* d_in, const int* in_sizes, int n_in,
                              void* d_out, int out_size, void* d_ws, size_t ws_size,
                              hipStream_t stream) {
    (void)in_sizes; (void)n_in; (void)out_size; (void)ws_size;
    const float* z   = (const float*)d_in[0];   // (32, 60, 8000) f32
    const float* emb = (const float*)d_in[1];   // (6, 40, 1024) f32
    float* out   = (float*)d_out;               // q_merge (15360000) ++ vq_loss (32)
    float* loss  = out + OUT_Q_ELEMS;
    float* bpack = (float*)d_ws;                // 245760 floats (0.94 MB), 16B aligned
    float* e2    = bpack + BP_TOTAL;            // 6*1024 floats = 24 KB

    vq_prep_kernel<<<(BP_TOTAL + 255) / 256, 256, 0, stream>>>(emb, bpack, e2, loss);
    vq_main_kernel<<<NTILES / 8, 256, 0, stream>>>(z, emb, bpack, e2, out, loss);
}